// MultiGrid_37486474559588
// MI455X (gfx1250) — compile-verified
//
#include <hip/hip_runtime.h>

// 4-byte-aligned float pair: lets us gather the (x0, x0+1) corner pair with a
// single global_load_b64 even when the element offset is odd (DWORD-aligned).
struct __attribute__((aligned(4))) FPair { float lo, hi; };

__device__ __forceinline__ float ssmooth(float t) { return t * t * (3.0f - 2.0f * t); }
__device__ __forceinline__ float lerpf(float a, float b, float t) {
    return __builtin_fmaf(t, b - a, a);
}

// Per-axis corner + weight, entirely in the float domain:
//   xf = clamp(floor(x), 0, W-2)   -> one v_floor_f32 + one v_med3_f32
//   t  = x - xf                    (exact in f32; t in [0,1] by construction:
//                                   x in [0, W-1], xf = min(floor(x), W-2))
// Clamping the low corner to W-2 instead of W-1 keeps the x-pair load
// contiguous; at the top edge t==1 and smoothstep(1)==1 reproduces the
// boundary value exactly, matching the reference's x0=W-1,t=0 case.
__device__ __forceinline__ void axis_corner(float x, int hi, int& i0, float& t) {
    float xf = __builtin_fminf(__builtin_fmaxf(floorf(x), 0.0f), (float)hi);
    i0 = (int)xf;
    t  = ssmooth(x - xf);
}

// Trilinear smoothstep sampling of one CxDxHxW volume (C=4) at one point.
template <int D, int H, int W>
__device__ __forceinline__ void sample_vol(const float* __restrict__ vol,
                                           float gx, float gy, float gz,
                                           float r[4]) {
    constexpr int HW  = H * W;
    constexpr int DHW = D * HW;

    float x = (gx + 1.0f) * (0.5f * (float)(W - 1));
    float y = (gy + 1.0f) * (0.5f * (float)(H - 1));
    float z = (gz + 1.0f) * (0.5f * (float)(D - 1));

    int x0, y0, z0;
    float tx, ty, tz;
    axis_corner(x, W - 2, x0, tx);
    axis_corner(y, H - 2, y0, ty);
    axis_corner(z, D - 2, z0, tz);

    // Base element offsets for the four (z,y) corner rows; x pair is contiguous.
    int b00 = (z0 * H + y0) * W + x0;  // (z0, y0)
    int b01 = b00 + W;                 // (z0, y1)
    int b10 = b00 + HW;                // (z1, y0)
    int b11 = b10 + W;                 // (z1, y1)

#pragma unroll
    for (int c = 0; c < 4; ++c) {
        const float* vc = vol + c * DHW;
        // Four b64 gathers per channel; default (RT) temporal hint so corner
        // data stays resident in the 192 MB L2 (all three volumes ~149 MB fit).
        FPair v00 = *(const FPair*)(vc + b00);
        FPair v01 = *(const FPair*)(vc + b01);
        FPair v10 = *(const FPair*)(vc + b10);
        FPair v11 = *(const FPair*)(vc + b11);

        float c00 = lerpf(v00.lo, v00.hi, tx);
        float c01 = lerpf(v01.lo, v01.hi, tx);
        float c10 = lerpf(v10.lo, v10.hi, tx);
        float c11 = lerpf(v11.lo, v11.hi, tx);
        float c0  = lerpf(c00, c01, ty);
        float c1  = lerpf(c10, c11, ty);
        r[c]      = lerpf(c0, c1, tz);
    }
}

__global__ __launch_bounds__(256) void multigrid_sample_kernel(
    const float* __restrict__ grid,
    const float* __restrict__ vol0,
    const float* __restrict__ vol1,
    const float* __restrict__ vol2,
    float* __restrict__ out, int N) {
    int n = blockIdx.x * blockDim.x + threadIdx.x;
    if (n >= N) return;

    // Grid is streamed exactly once: non-temporal loads (one b96 NT per lane)
    // keep it from evicting the L2-resident volumes.
    const float* g = grid + 3ll * n;
    float gx = __builtin_nontemporal_load(g + 0);
    float gy = __builtin_nontemporal_load(g + 1);
    float gz = __builtin_nontemporal_load(g + 2);

    float r0[4], r1[4], r2[4];
    sample_vol< 64,  64,  64>(vol0, gx, gy, gz, r0);
    sample_vol<128, 128, 128>(vol1, gx, gy, gz, r1);
    sample_vol<192, 192, 192>(vol2, gx, gy, gz, r2);

    // Output layout [1, 12, 1, 1, N] -> out[c*N + n]; lane-consecutive b32
    // stores are fully coalesced per channel plane; write-once stream, so
    // non-temporal stores (th:NT) avoid polluting L2.
    float* o = out + n;
#pragma unroll
    for (int c = 0; c < 4; ++c)
        __builtin_nontemporal_store(r0[c], o + (long long)c * N);
#pragma unroll
    for (int c = 0; c < 4; ++c)
        __builtin_nontemporal_store(r1[c], o + (long long)(4 + c) * N);
#pragma unroll
    for (int c = 0; c < 4; ++c)
        __builtin_nontemporal_store(r2[c], o + (long long)(8 + c) * N);
}

extern "C" void kernel_launch(void* const* d_in, const int* in_sizes, int n_in,
                              void* d_out, int out_size, void* d_ws, size_t ws_size,
                              hipStream_t stream) {
    const float* grid = (const float*)d_in[0];
    const float* vol0 = (const float*)d_in[1];
    const float* vol1 = (const float*)d_in[2];
    const float* vol2 = (const float*)d_in[3];
    float* out = (float*)d_out;

    int N = in_sizes[0] / 3;  // grid is [1,1,1,N,3]
    int threads = 256;        // 8 wave32s per block
    int blocks = (N + threads - 1) / threads;
    multigrid_sample_kernel<<<blocks, threads, 0, stream>>>(grid, vol0, vol1, vol2, out, N);
}